// GINEncoder_84482006712591
// MI455X (gfx1250) — compile-verified
//
#include <hip/hip_runtime.h>

#define D 128
#define WT_STRIDE 132   // pad so wave32 ds_load_b64 B-fragments are bank-conflict-free

typedef float v2f __attribute__((ext_vector_type(2)));
typedef float v8f __attribute__((ext_vector_type(8)));

// ---------------------------------------------------------------------------
// Row copy: h = x  (vectorized float4)
// ---------------------------------------------------------------------------
__global__ void gin_copy_rows(const float4* __restrict__ src,
                              float4* __restrict__ dst, long long n4) {
  long long i = (long long)blockIdx.x * blockDim.x + threadIdx.x;
  if (i < n4) dst[i] = src[i];
}

// ---------------------------------------------------------------------------
// Edge scatter-add: h[dst,:] += x[src,:].  One wave per edge, each lane owns a
// float4 chunk of the 128-wide row -> 4 native global_atomic_add_f32 per lane.
// ---------------------------------------------------------------------------
__global__ void gin_scatter_add(const float* __restrict__ x,
                                const long long* __restrict__ ei,
                                float* __restrict__ h, long long E) {
  long long tid = (long long)blockIdx.x * blockDim.x + threadIdx.x;
  long long e = tid >> 5;
  int lane = (int)(tid & 31);
  if (e >= E) return;
  long long s = ei[e];
  long long d = ei[E + e];
  const float4 v = *(const float4*)(x + s * D + lane * 4);
  float* hp = h + d * D + lane * 4;
  unsafeAtomicAdd(hp + 0, v.x);
  unsafeAtomicAdd(hp + 1, v.y);
  unsafeAtomicAdd(hp + 2, v.z);
  unsafeAtomicAdd(hp + 3, v.w);
}

// ---------------------------------------------------------------------------
// O = relu(M @ W + bias), M:[nrows,128], W:[128,128] row-major, via
// V_WMMA_F32_16X16X4_F32.
//
// Block = 256 threads (8 waves), 128 rows per block.  W is staged TRANSPOSED
// into LDS once per block (WT[n][k], row stride 132 floats) so each B
// fragment is a single bank-conflict-free ds_load_b64 (compiler pairs them
// into ds_load_2addr_b64), shared by all 8 waves across the whole K loop.
// Each wave owns one 16-row strip and all 8 column tiles (8 independent v8f
// accumulation chains feeding the XDL pipe).
//
// CDNA5 fragment layouts (ISA 7.12.2):
//   A 16x4 f32:  lanes 0-15 M=lane {K=k0,k0+1}; lanes 16-31 M=lane-16 {k0+2,k0+3}
//   B 4x16 f32:  lanes 0-15 N=lane {K=k0,k0+1}; lanes 16-31 N=lane-16 {k0+2,k0+3}
//   C/D 16x16:   VGPR r -> M=r (lanes 0-15) / M=r+8 (lanes 16-31), N=lane%16
// EXEC is all ones through the WMMA region (row clamp, no divergence).  The
// epilogue takes a wave-uniform branch: unpredicated stores for full strips,
// per-row predication only in the single partial tail block.
// ---------------------------------------------------------------------------
__global__ __launch_bounds__(256) void gin_mlp_gemm_relu(
    const float* __restrict__ M, const float* __restrict__ W,
    const float* __restrict__ bias, float* __restrict__ O, int nrows) {
  extern __shared__ float wt[];  // [128][WT_STRIDE] = W transposed, padded

  const int lane = threadIdx.x & 31;
  const int wave = threadIdx.x >> 5;
  const int lr   = lane & 15;
  const int half = lane >> 4;
  const int row_base = blockIdx.x * 128 + wave * 16;

  // ---- stage W transposed into LDS (coalesced float4 global reads) ----
  for (int i = threadIdx.x; i < (D * D) / 4; i += 256) {
    const float4 w = ((const float4*)W)[i];
    const int j = i * 4;        // linear index into row-major W
    const int k = j >> 7;       // W row    (K index)
    const int n = j & (D - 1);  // W column (N index)
    wt[(n + 0) * WT_STRIDE + k] = w.x;
    wt[(n + 1) * WT_STRIDE + k] = w.y;
    wt[(n + 2) * WT_STRIDE + k] = w.z;
    wt[(n + 3) * WT_STRIDE + k] = w.w;
  }
  __syncthreads();

  v8f acc[8];
#pragma unroll
  for (int i = 0; i < 8; ++i) acc[i] = (v8f)(0.0f);

  int mA = row_base + lr;
  if (mA > nrows - 1) mA = nrows - 1;   // clamp, keep EXEC uniform
  const float* arow = M + (size_t)mA * D;
  const float* wrow = wt + (size_t)lr * WT_STRIDE;  // WT[lr + 16*nt][*]

#pragma unroll 2
  for (int k0 = 0; k0 < D; k0 += 4) {
    const int ka = k0 + 2 * half;
    v2f a = *(const v2f*)(arow + ka);  // two consecutive K of this lane's row
#pragma unroll
    for (int nt = 0; nt < 8; ++nt) {
      // B fragment: single ds_load_b64, 64-bank conflict-free (stride 132)
      v2f b = *(const v2f*)(wrow + nt * 16 * WT_STRIDE + ka);
      acc[nt] = __builtin_amdgcn_wmma_f32_16x16x4_f32(
          false, a, false, b, (short)0, acc[nt], false, false);
    }
  }

  // ---- bias + relu + store ----
  if (row_base + 16 <= nrows) {
    // fast path (all blocks except the final partial one): no predication
#pragma unroll
    for (int nt = 0; nt < 8; ++nt) {
      const int n = nt * 16 + lr;
      const float bv = bias[n];
      float* op = O + (size_t)(row_base + 8 * half) * D + n;
#pragma unroll
      for (int r = 0; r < 8; ++r) {
        float v = acc[nt][r] + bv;
        op[(size_t)r * D] = v > 0.0f ? v : 0.0f;
      }
    }
  } else {
    // tail path: predicate per row
#pragma unroll
    for (int nt = 0; nt < 8; ++nt) {
      const int n = nt * 16 + lr;
      const float bv = bias[n];
#pragma unroll
      for (int r = 0; r < 8; ++r) {
        const int m = row_base + r + 8 * half;
        if (m < nrows) {
          float v = acc[nt][r] + bv;
          O[(size_t)m * D + n] = v > 0.0f ? v : 0.0f;
        }
      }
    }
  }
}

// ---------------------------------------------------------------------------
// Launch: 2 GIN layers.  bufA = d_ws, bufB = d_out (ping-pong, 51.2 MB each).
// ---------------------------------------------------------------------------
extern "C" void kernel_launch(void* const* d_in, const int* in_sizes, int n_in,
                              void* d_out, int out_size, void* d_ws, size_t ws_size,
                              hipStream_t stream) {
  const float*     x   = (const float*)d_in[0];
  const long long* ei  = (const long long*)d_in[1];
  // d_in[2] = batch (unused)
  const float* W1a = (const float*)d_in[3];
  const float* b1a = (const float*)d_in[4];
  const float* W1b = (const float*)d_in[5];
  const float* b1b = (const float*)d_in[6];
  const float* W2a = (const float*)d_in[7];
  const float* b2a = (const float*)d_in[8];
  const float* W2b = (const float*)d_in[9];
  const float* b2b = (const float*)d_in[10];

  const int       N  = in_sizes[0] / D;
  const long long E  = (long long)in_sizes[1] / 2;
  float* bufA = (float*)d_ws;   // N*128 floats
  float* bufB = (float*)d_out;  // N*128 floats

  const long long n4 = (long long)N * (D / 4);
  const int copyGrid = (int)((n4 + 255) / 256);
  const int scatGrid = (int)((E * 32 + 255) / 256);
  const int gemmGrid = (N + 127) / 128;
  const size_t ldsBytes = (size_t)D * WT_STRIDE * sizeof(float);  // 67.6 KB

  // ---- layer 1 ----
  // h0 = x + segment_sum(x[src] -> dst)
  gin_copy_rows<<<copyGrid, 256, 0, stream>>>((const float4*)x, (float4*)bufA, n4);
  gin_scatter_add<<<scatGrid, 256, 0, stream>>>(x, ei, bufA, E);
  // t1 = relu(h0 @ W1a + b1a); h1 = relu(t1 @ W1b + b1b)
  gin_mlp_gemm_relu<<<gemmGrid, 256, ldsBytes, stream>>>(bufA, W1a, b1a, bufB, N);
  gin_mlp_gemm_relu<<<gemmGrid, 256, ldsBytes, stream>>>(bufB, W1b, b1b, bufA, N);

  // ---- layer 2 ----
  // g = h1 + segment_sum(h1[src] -> dst)
  gin_copy_rows<<<copyGrid, 256, 0, stream>>>((const float4*)bufA, (float4*)bufB, n4);
  gin_scatter_add<<<scatGrid, 256, 0, stream>>>(bufA, ei, bufB, E);
  // t2 = relu(g @ W2a + b2a); out = relu(t2 @ W2b + b2b)
  gin_mlp_gemm_relu<<<gemmGrid, 256, ldsBytes, stream>>>(bufB, W2a, b2a, bufA, N);
  gin_mlp_gemm_relu<<<gemmGrid, 256, ldsBytes, stream>>>(bufA, W2b, b2b, (float*)d_out, N);
}